// GraphSAGELayer_47107201303323
// MI455X (gfx1250) — compile-verified
//
#include <hip/hip_runtime.h>
#include <hip/hip_bf16.h>

#define N_NODES 100000
#define D_FEAT  128
#define MTILES  5           // M-tiles per wave in the GEMM (6250 = 1250 * 5)

typedef float v2f __attribute__((ext_vector_type(2)));
typedef float v8f __attribute__((ext_vector_type(8)));

// ---------------------------------------------------------------------------
// Kernel 1: zero the aggregation accumulator (ws) — float4 stores.
// ---------------------------------------------------------------------------
__global__ void sage_zero_kernel(float4* __restrict__ p, int n4) {
    int i = blockIdx.x * blockDim.x + threadIdx.x;
    if (i < n4) {
        p[i] = make_float4(0.f, 0.f, 0.f, 0.f);
    }
}

// ---------------------------------------------------------------------------
// Kernel 2: edge scatter-add.  One wave (32 lanes) per edge; each lane owns
// 4 consecutive features -> one coalesced 16B gather of h[src], then 4
// global_atomic_add_f32 into ah[dst].  h (51 MB) and ah (51 MB) both fit in
// the 192 MB L2, so the random gather/atomic traffic stays on-chip.
// ---------------------------------------------------------------------------
__global__ void sage_scatter_kernel(const float* __restrict__ h,
                                    const int* __restrict__ edge_src,
                                    const int* __restrict__ edge_dst,
                                    float* __restrict__ ah,
                                    int n_edges) {
    int t    = blockIdx.x * blockDim.x + threadIdx.x;
    int e    = t >> 5;          // edge index (wave per edge)
    int lane = t & 31;          // lane -> feature group
    if (e >= n_edges) return;

    int s = edge_src[e];
    int d = edge_dst[e];

    const float4 v = *(const float4*)(h + (size_t)s * D_FEAT + lane * 4);
    float* o = ah + (size_t)d * D_FEAT + lane * 4;
    atomicAdd(o + 0, v.x);
    atomicAdd(o + 1, v.y);
    atomicAdd(o + 2, v.z);
    atomicAdd(o + 3, v.w);
}

// ---------------------------------------------------------------------------
// Kernel 3: out = relu(ah @ W^T + b) via V_WMMA_F32_16X16X4_F32 (fp32 exact).
// Block = 256 threads = 8 waves.  Wave w owns N-tile w (cols w*16..w*16+15)
// and processes MTILES consecutive 16-row M-tiles, holding all 32 B-matrix
// fragments (64 VGPRs) in registers across the M loop.  This removes the
// dominant redundant W refetch (~400 MB -> ~80 MB of L2 traffic).
//
// Fragment layouts (ISA 7.12.2, wave32):
//   A (16x4 f32, 2 VGPRs):  lanes 0-15: row=lane,    K = {k0, k0+1}
//                           lanes 16-31: row=lane-16, K = {k0+2, k0+3}
//   B (4x16 f32, 2 VGPRs):  mirrored — lane selects column N, lane-half
//                           selects K pair.  B[k][n] = W[n][k] (row-major W).
//   C/D (16x16 f32, 8 VGPRs): VGPR i: lanes 0-15 -> M=i, N=lane;
//                             lanes 16-31 -> M=i+8, N=lane-16.
// ---------------------------------------------------------------------------
__global__ void sage_gemm_relu_kernel(const float* __restrict__ ah,
                                      const float* __restrict__ W,
                                      const float* __restrict__ bias,
                                      float* __restrict__ out) {
    const int lane = threadIdx.x & 31;
    const int wave = threadIdx.x >> 5;        // 0..7 -> N tile
    const int col0 = wave * 16;               // N tile base
    const int m0   = blockIdx.x * (16 * MTILES);

    const int half = lane >> 4;               // 0: K pair {0,1}; 1: {2,3}
    const int l15  = lane & 15;

    // Preload all B fragments for this wave's N-tile (stay in VGPRs).
    const float* wrow = W + (size_t)(col0 + l15) * D_FEAT + half * 2;
    v2f bfrag[D_FEAT / 4];
#pragma unroll
    for (int kk = 0; kk < D_FEAT / 4; ++kk) {
        bfrag[kk].x = wrow[4 * kk];
        bfrag[kk].y = wrow[4 * kk + 1];
    }
    const float bv = bias[col0 + l15];

    for (int mt = 0; mt < MTILES; ++mt) {
        const int row0 = m0 + mt * 16;
        const float* arow = ah + (size_t)(row0 + l15) * D_FEAT + half * 2;

        v8f acc = {};
#pragma unroll
        for (int kk = 0; kk < D_FEAT / 4; ++kk) {
            v2f a;
            a.x = arow[4 * kk];
            a.y = arow[4 * kk + 1];
            // 8 args: (neg_a, A, neg_b, B, c_mod, C, reuse_a, reuse_b)
            acc = __builtin_amdgcn_wmma_f32_16x16x4_f32(
                false, a, false, bfrag[kk], (short)0, acc, false, false);
        }

#pragma unroll
        for (int i = 0; i < 8; ++i) {
            const int r = i + half * 8;
            float v = acc[i] + bv;
            v = v > 0.f ? v : 0.f;
            // Output is write-once / never re-read: nontemporal store keeps
            // the L2-resident h/ah working set from being evicted.
            __builtin_nontemporal_store(
                v, &out[(size_t)(row0 + r) * D_FEAT + col0 + l15]);
        }
    }
}

// ---------------------------------------------------------------------------
// Launcher.  Inputs: h [N,128] f32, edge_src [E] i32, edge_dst [E] i32,
// W [128,128] f32, b [128] f32.  Output: [N,128] f32.  ws holds ah [N,128].
// ---------------------------------------------------------------------------
extern "C" void kernel_launch(void* const* d_in, const int* in_sizes, int n_in,
                              void* d_out, int out_size, void* d_ws, size_t ws_size,
                              hipStream_t stream) {
    const float* h        = (const float*)d_in[0];
    const int*   edge_src = (const int*)d_in[1];
    const int*   edge_dst = (const int*)d_in[2];
    const float* W        = (const float*)d_in[3];
    const float* bias     = (const float*)d_in[4];
    float*       out      = (float*)d_out;
    float*       ah       = (float*)d_ws;          // [N_NODES, D_FEAT] accumulator

    const int n_edges = in_sizes[1];

    // 1) zero accumulator
    {
        const int n4 = N_NODES * D_FEAT / 4;
        const int blocks = (n4 + 255) / 256;
        sage_zero_kernel<<<blocks, 256, 0, stream>>>((float4*)ah, n4);
    }

    // 2) scatter-add over edges (wave per edge)
    {
        const long long threads = (long long)n_edges * 32;
        const int blocks = (int)((threads + 255) / 256);
        sage_scatter_kernel<<<blocks, 256, 0, stream>>>(h, edge_src, edge_dst, ah, n_edges);
    }

    // 3) WMMA GEMM + bias + ReLU (register-resident B across 5 M-tiles)
    {
        const int mblocks = N_NODES / (16 * MTILES);  // 1250 (exact)
        sage_gemm_relu_kernel<<<mblocks, 256, 0, stream>>>(ah, W, bias, out);
    }
}